// IF_57664230916743
// MI455X (gfx1250) — compile-verified
//
#include <hip/hip_runtime.h>

// IF spiking-neuron forward scan (T=8 timesteps, sequential per spatial site).
// Pure HBM-streaming kernel: 128-bit non-temporal loads/stores, 32-bit
// addressing (total footprint 2^27 B), scan carried entirely in registers.
// Roofline: 256 MiB single-pass traffic / 23.3 TB/s ≈ 11.5 us; no matrix ops
// exist in this op, so the vmem path (not WMMA) is the relevant CDNA5 feature.

typedef float f32x4 __attribute__((ext_vector_type(4)));

static constexpr int T = 8;

__global__ __launch_bounds__(256) void if_scan_vec4(
    const float* __restrict__ x,
    const float* __restrict__ thresh,
    float* __restrict__ out,
    unsigned int nvec)        // n_space / 4 == vec4 stride between timesteps
{
    const unsigned int i = blockIdx.x * 256u + threadIdx.x;   // vec4 column id
    if (i >= nvec) return;

    const float thre  = thresh[0];            // uniform -> scalar load
    const float nthre = -thre;

    const f32x4* __restrict__ xp = (const f32x4*)x;
    f32x4* __restrict__ op = (f32x4*)out;

    // Issue all 8 timestep loads first: 8 outstanding LOADcnt entries per
    // wave (global_load_b128 th:NT) for maximum memory-level parallelism.
    f32x4 xv[T];
#pragma unroll
    for (int t = 0; t < T; ++t)
        xv[t] = __builtin_nontemporal_load(xp + (size_t)(i + (unsigned)t * nvec));

    f32x4 mem  = { 0.5f * thre, 0.5f * thre, 0.5f * thre, 0.5f * thre };
    f32x4 flag = { 0.0f, 0.0f, 0.0f, 0.0f };

#pragma unroll
    for (int t = 0; t < T; ++t) {
        const f32x4 m = mem + xv[t];
        f32x4 spike;
#pragma unroll
        for (int k = 0; k < 4; ++k) {
            const float mm = m[k];
            // zif(mem - thre): fires +thre when mem >= thre
            const float pos = (mm >= thre) ? thre : 0.0f;
            // negative spike gated by accumulated flag > 0
            const float neg = ((mm <= nthre) && (flag[k] > 0.0f)) ? nthre : 0.0f;
            spike[k] = pos + neg;
        }
        mem  = m - spike;
        flag = flag + spike;
        __builtin_nontemporal_store(spike, op + (size_t)(i + (unsigned)t * nvec));
    }
}

// Remainder kernel: only launched when n_space % 4 != 0 (never for the
// reference shapes). Handles the last `rem` scalar columns with one thread
// each, keeping the hot kernel free of tail code.
__global__ void if_scan_tail(
    const float* __restrict__ x,
    const float* __restrict__ thresh,
    float* __restrict__ out,
    unsigned int n_space,
    unsigned int rem)
{
    const unsigned int r = threadIdx.x;
    if (r >= rem) return;
    const unsigned int e = n_space - rem + r;

    const float thre  = thresh[0];
    const float nthre = -thre;

    float mem  = 0.5f * thre;
    float flag = 0.0f;
    for (int t = 0; t < T; ++t) {
        const float m   = mem + x[(size_t)((unsigned)t * n_space + e)];
        const float pos = (m >= thre) ? thre : 0.0f;
        const float neg = ((m <= nthre) && (flag > 0.0f)) ? nthre : 0.0f;
        const float spike = pos + neg;
        mem  = m - spike;
        flag = flag + spike;
        out[(size_t)((unsigned)t * n_space + e)] = spike;
    }
}

extern "C" void kernel_launch(void* const* d_in, const int* in_sizes, int n_in,
                              void* d_out, int out_size, void* d_ws, size_t ws_size,
                              hipStream_t stream) {
    const float* x      = (const float*)d_in[0];   // [T*B, C, H, W] f32
    const float* thresh = (const float*)d_in[1];   // [1] f32
    float* out          = (float*)d_out;           // same shape as x

    const unsigned int total   = (unsigned int)in_sizes[0];
    const unsigned int n_space = total / (unsigned int)T;   // B*C*H*W
    const unsigned int nvec    = n_space >> 2;
    const unsigned int rem     = n_space & 3u;

    const int block = 256;                                  // 8 wave32 waves
    const unsigned int grid = (nvec + (unsigned)block - 1u) / (unsigned)block;

    if_scan_vec4<<<grid, block, 0, stream>>>(x, thresh, out, nvec);

    if (rem != 0u) {   // host-side constant per shape: deterministic
        if_scan_tail<<<1, 32, 0, stream>>>(x, thresh, out, n_space, rem);
    }
}